// BigramKLLoss_84421877170331
// MI455X (gfx1250) — compile-verified
//
#include <hip/hip_runtime.h>
#include <hip/hip_bf16.h>
#include <math.h>

// Problem constants (match reference: B=4, T=1024, V=32000, K=50000)
constexpr int Bdim = 4;
constexpr int Tdim = 1024;
constexpr int Vdim = 32000;
constexpr int Kdim = 50000;
constexpr int NBT  = Bdim * (Tdim - 1);           // 4092 (b,t) row pairs
constexpr float EPST = 1e-8f;
constexpr float EPSM = 1e-12f;

#define BLOCK   1024                              // 32 waves -> 8 per SIMD32
#define ACCS    8
#define KCHUNK  (BLOCK * ACCS)                    // 8192 k per block
#define NKC     ((Kdim + KCHUNK - 1) / KCHUNK)    // 7
#define BT_CHUNK 128
#define NBTC    ((NBT + BT_CHUNK - 1) / BT_CHUNK) // 32
#define FBLOCK  256                               // finalize block

#if defined(__has_builtin)
#if __has_builtin(__builtin_amdgcn_global_load_async_to_lds_b128) && \
    __has_builtin(__builtin_amdgcn_s_wait_asynccnt)
#define HAVE_ASYNC_LDS 1
#endif
#endif

typedef int v4i __attribute__((vector_size(16)));
typedef __attribute__((address_space(1))) v4i* gas_v4i_ptr;  // global (AS1)
typedef __attribute__((address_space(3))) v4i* las_v4i_ptr;  // LDS (AS3)

// Copy one 128KB vocab row (32000 f32 = 8000 x b128) global -> LDS (async).
__device__ __forceinline__ void copy_row(const float* __restrict__ g,
                                         float* l, int tid) {
#ifdef HAVE_ASYNC_LDS
  for (int c = tid; c < Vdim / 4; c += BLOCK) {
    __builtin_amdgcn_global_load_async_to_lds_b128(
        (gas_v4i_ptr)(g + 4 * c),
        (las_v4i_ptr)(l + 4 * c),
        /*offset=*/0, /*cpol=*/0);
  }
#else
  for (int c = tid; c < Vdim / 4; c += BLOCK) {
    reinterpret_cast<float4*>(l)[c] = reinterpret_cast<const float4*>(g)[c];
  }
#endif
}

__device__ __forceinline__ void wait_async_copies() {
#ifdef HAVE_ASYNC_LDS
  __builtin_amdgcn_s_wait_asynccnt(0);
#endif
}

__global__ void zero_f32(float* __restrict__ p, int n) {
  int i = blockIdx.x * blockDim.x + threadIdx.x;
  if (i < n) p[i] = 0.0f;
}

// Phase 1: stream row pairs through a 2-slot LDS buffer; overlap the async
// copy of the next Q row with the P-side LDS gathers.
__global__ __launch_bounds__(BLOCK) void bigram_accum(
    const float* __restrict__ probs, const unsigned char* __restrict__ mask,
    const int* __restrict__ pairs, float* __restrict__ topk_sum) {
  extern __shared__ float lds[];  // 2 * Vdim floats (250 KB)
  const int tid = threadIdx.x;
  const int kc  = blockIdx.x;
  const int btc = blockIdx.y;
  const int k0  = kc * KCHUNK;

  int   ai[ACCS], bi[ACCS];
  float acc[ACCS];
#pragma unroll
  for (int i = 0; i < ACCS; ++i) {
    const int k = k0 + tid + i * BLOCK;
    acc[i] = 0.0f;
    if (k < Kdim) { ai[i] = pairs[2 * k]; bi[i] = pairs[2 * k + 1]; }
    else          { ai[i] = 0;            bi[i] = 0; }
  }

  const int bt0 = btc * BT_CHUNK;
  const int bt1 = (bt0 + BT_CHUNK < NBT) ? (bt0 + BT_CHUNK) : NBT;
  int curP = -1, curQ = -1, slotP = 0, slotQ = 1;

  for (int bt = bt0; bt < bt1; ++bt) {
    const int bb = bt / (Tdim - 1);
    const int tt = bt - bb * (Tdim - 1);
    // pair_mask (uniform across the block: every thread reads the same bytes)
    if (!(mask[bb * Tdim + tt] && mask[bb * Tdim + tt + 1])) continue;

    const int rP = bb * Tdim + tt;
    const int rQ = rP + 1;
    if (curP != rP && curQ == rP) {  // previous Q row becomes new P row
      int s = slotP; slotP = slotQ; slotQ = s;
      int c = curP;  curP  = curQ;  curQ  = c;
    }
    const bool needP = (curP != rP);
    const bool needQ = (curQ != rQ);
    const float* __restrict__ lp = lds + slotP * Vdim;
    const float* __restrict__ lq = lds + slotQ * Vdim;

    if (needP) {
      // Cold start (or post-mask-gap): load both rows, no overlap possible.
      __syncthreads();  // WAR: prior gathers done before overwriting slots
      copy_row(probs + (size_t)rP * Vdim, lds + slotP * Vdim, tid); curP = rP;
      if (needQ) { copy_row(probs + (size_t)rQ * Vdim, lds + slotQ * Vdim, tid); curQ = rQ; }
      wait_async_copies();
      __syncthreads();
#pragma unroll
      for (int i = 0; i < ACCS; ++i)
        acc[i] = fmaf(lp[ai[i]], lq[bi[i]], acc[i]);
    } else if (needQ) {
      // Steady state: P is valid (slot swap); stream Q while gathering P.
      __syncthreads();  // WAR: prior gathers done before overwriting slotQ
      copy_row(probs + (size_t)rQ * Vdim, lds + slotQ * Vdim, tid); curQ = rQ;
      float pa[ACCS];
#pragma unroll
      for (int i = 0; i < ACCS; ++i)
        pa[i] = lp[ai[i]];               // overlaps with the async copy
      wait_async_copies();
      __syncthreads();
#pragma unroll
      for (int i = 0; i < ACCS; ++i)
        acc[i] = fmaf(pa[i], lq[bi[i]], acc[i]);
    } else {
      // Both rows already resident.
#pragma unroll
      for (int i = 0; i < ACCS; ++i)
        acc[i] = fmaf(lp[ai[i]], lq[bi[i]], acc[i]);
    }
  }

#pragma unroll
  for (int i = 0; i < ACCS; ++i) {
    const int k = k0 + tid + i * BLOCK;
    if (k < Kdim) atomicAdd(&topk_sum[k], acc[i]);
  }
}

// Phase 2: n_pairs + clamped KL reduction -> scalar.
__global__ __launch_bounds__(FBLOCK) void bigram_finalize(
    const float* __restrict__ topk_sum, const float* __restrict__ target_probs,
    const float* __restrict__ target_oov, const unsigned char* __restrict__ mask,
    float* __restrict__ out) {
  __shared__ double red0[FBLOCK];
  __shared__ double red1[FBLOCK];
  const int tid = threadIdx.x;

  // n_pairs = sum of pair_mask
  float cnt = 0.0f;
  for (int idx = tid; idx < NBT; idx += FBLOCK) {
    const int bb = idx / (Tdim - 1);
    const int tt = idx - bb * (Tdim - 1);
    cnt += (mask[bb * Tdim + tt] && mask[bb * Tdim + tt + 1]) ? 1.0f : 0.0f;
  }
  red0[tid] = (double)cnt;
  __syncthreads();
  for (int s = FBLOCK / 2; s > 0; s >>= 1) {
    if (tid < s) red0[tid] += red0[tid + s];
    __syncthreads();
  }
  const float n = fmaxf((float)red0[0], 1.0f);
  __syncthreads();

  double s_mt = 0.0, s_kl = 0.0;
  for (int k = tid; k < Kdim; k += FBLOCK) {
    const float mt = fmaxf(topk_sum[k] / n, EPSM);
    const float tg = fmaxf(target_probs[k], EPST);
    s_mt += (double)mt;
    s_kl += (double)mt * (double)(logf(mt) - logf(tg));
  }
  red0[tid] = s_mt;
  red1[tid] = s_kl;
  __syncthreads();
  for (int s = FBLOCK / 2; s > 0; s >>= 1) {
    if (tid < s) { red0[tid] += red0[tid + s]; red1[tid] += red1[tid + s]; }
    __syncthreads();
  }
  if (tid == 0) {
    float moov = 1.0f - (float)red0[0];
    moov = fminf(fmaxf(moov, EPSM), 1.0f - EPST);
    const float toov  = fmaxf(target_oov[0], EPST);
    const float kloov = moov * (logf(moov) - logf(toov));
    out[0] = (float)red1[0] + kloov;
  }
}

extern "C" void kernel_launch(void* const* d_in, const int* in_sizes, int n_in,
                              void* d_out, int out_size, void* d_ws, size_t ws_size,
                              hipStream_t stream) {
  const float*         probs        = (const float*)d_in[0];
  const float*         target_probs = (const float*)d_in[1];
  const float*         target_oov   = (const float*)d_in[2];
  const unsigned char* mask         = (const unsigned char*)d_in[3];
  const int*           pairs        = (const int*)d_in[4];
  float* out  = (float*)d_out;
  float* topk = (float*)d_ws;  // Kdim floats (200 KB)

  zero_f32<<<(Kdim + 255) / 256, 256, 0, stream>>>(topk, Kdim);

  dim3 grid(NKC, NBTC);
  const size_t shmem = 2 * (size_t)Vdim * sizeof(float);  // 250 KB
  bigram_accum<<<grid, BLOCK, shmem, stream>>>(probs, mask, pairs, topk);

  bigram_finalize<<<1, FBLOCK, 0, stream>>>(topk, target_probs, target_oov, mask, out);
}